// SOENModel_21543555956738
// MI455X (gfx1250) — compile-verified
//
#include <hip/hip_runtime.h>
#include <math.h>

#define NN   4096
#define NB   32
#define NIN  1024
#define NHID 2944
#define NOUT 128
#define DT_F 0.05f

typedef float v2f __attribute__((ext_vector_type(2)));
typedef float v8f __attribute__((ext_vector_type(8)));

// ---------------- zero init (s0 = 0) ----------------
__global__ void soen_zero_kernel(float* __restrict__ p, int n) {
    int i = blockIdx.x * blockDim.x + threadIdx.x;
    if (i < n) p[i] = 0.0f;
}

// ---------------- Jm = J * mask (once) ----------------
__global__ void soen_maskmul_kernel(const float4* __restrict__ J,
                                    const float4* __restrict__ M,
                                    float4* __restrict__ Jm, int n4) {
    int i = blockIdx.x * blockDim.x + threadIdx.x;
    if (i < n4) {
        float4 j = J[i], m = M[i];
        Jm[i] = make_float4(j.x * m.x, j.y * m.y, j.z * m.z, j.w * m.w);
    }
}

// ---------------- one recurrence step ----------------
// Grid: 64 blocks (each owns a 64-wide column slab of phi), 256 threads = 8 waves.
// Wave w reduces K-range [w*512, (w+1)*512) with V_WMMA_F32_16X16X4_F32.
// Each wave keeps 8 accumulators: 2 batch tiles (b 0-15, 16-31) x 4 column tiles,
// so every A (state) fragment load feeds 4 WMMAs -> A-side L2 traffic /4.
// 64 KB LDS reduction across the 8 waves, then fused elementwise SOEN update.
template <bool USE_MASK>
__global__ __launch_bounds__(256)
void soen_step_kernel(const float* __restrict__ scur,
                      float* __restrict__ snxt,
                      const float* __restrict__ Jm,     // masked J (raw J if USE_MASK)
                      const float* __restrict__ mask,   // only if USE_MASK
                      const float* __restrict__ x,      // [32,1024]
                      const float* __restrict__ flux,   // [4096]
                      const float* __restrict__ gamma,  // [4096]
                      const float* __restrict__ tau,    // [4096]
                      const float* __restrict__ noise_t)// [32,4096] slice for this step
{
    __shared__ float lds[64 * 32 * 8];   // [w*8 + t*4 + tt][lane][reg] = 64 KB

    const int tid  = threadIdx.x;
    const int w    = tid >> 5;           // wave 0..7
    const int lane = tid & 31;
    const int ln   = lane & 15;          // row within fragment
    const int co   = (lane >> 4) << 1;   // K sub-offset: 0 or 2
    const int i0   = blockIdx.x << 6;    // 64-column slab base
    const int kb   = w << 9;             // this wave's K base (512 per wave)

    // B fragments: B[k,n] = Jm[i0 + tt*16 + n, k0+k]; tile offset tt*16*NN folds
    // into the 24-bit instruction immediate.
    const float* pB  = Jm   + (size_t)(i0 + ln) * NN + kb + co;
    const float* pM  = USE_MASK ? (mask + (size_t)(i0 + ln) * NN + kb + co) : nullptr;
    // A fragments: A[m,k] = s[b0+m, k0+k]
    const float* pA0 = scur + (size_t)(ln)      * NN + kb + co;  // batch rows 0-15
    const float* pA1 = scur + (size_t)(16 + ln) * NN + kb + co;  // batch rows 16-31

    v8f acc[2][4];
    #pragma unroll
    for (int t = 0; t < 2; ++t)
        #pragma unroll
        for (int tt = 0; tt < 4; ++tt)
            acc[t][tt] = (v8f){};

    #pragma unroll 4
    for (int kk = 0; kk < 512; kk += 4) {
        v2f a0 = *(const v2f*)(pA0 + kk);
        v2f a1 = *(const v2f*)(pA1 + kk);
        v2f b[4];
        #pragma unroll
        for (int tt = 0; tt < 4; ++tt) {
            b[tt] = *(const v2f*)(pB + (size_t)tt * 16 * NN + kk);
            if (USE_MASK) {
                v2f m = *(const v2f*)(pM + (size_t)tt * 16 * NN + kk);
                b[tt].x *= m.x;
                b[tt].y *= m.y;
            }
        }
        #pragma unroll
        for (int tt = 0; tt < 4; ++tt) {
            acc[0][tt] = __builtin_amdgcn_wmma_f32_16x16x4_f32(
                false, a0, false, b[tt], (short)0, acc[0][tt], false, false);
            acc[1][tt] = __builtin_amdgcn_wmma_f32_16x16x4_f32(
                false, a1, false, b[tt], (short)0, acc[1][tt], false, false);
        }
    }

    // stash partials: [w*8 + t*4 + tt][lane][reg] -> contiguous 8 floats per lane
    #pragma unroll
    for (int t = 0; t < 2; ++t) {
        #pragma unroll
        for (int tt = 0; tt < 4; ++tt) {
            float* d = &lds[(((w << 3) + (t << 2) + tt) * 32 + lane) << 3];
            #pragma unroll
            for (int r = 0; r < 8; ++r) d[r] = acc[t][tt][r];
        }
    }
    __syncthreads();

    // reduce 8 K-partials and apply SOEN update for 2048 elements (32 b x 64 i)
    for (int e = tid; e < 2048; e += 256) {
        const int b   = e >> 6;            // batch row 0..31
        const int col = e & 63;            // column within slab
        const int tt  = col >> 4;          // column tile
        const int n   = col & 15;          // column within 16x16 D tile
        const int t   = b >> 4;            // accumulator batch-tile
        const int m   = b & 15;            // row within D tile
        const int r   = m & 7;             // VGPR index in D layout
        const int l   = n + ((m >> 3) << 4); // lane in D layout

        float phi = 0.0f;
        #pragma unroll
        for (int ww = 0; ww < 8; ++ww)
            phi += lds[((((ww << 3) + (t << 2) + tt) * 32 + l) << 3) + r];

        const int i = i0 + col;
        phi += flux[i];
        if (i < NIN) phi += x[b * NIN + i];
        phi += noise_t[(size_t)b * NN + i];
        phi = fminf(fmaxf(phi, -0.5f), 0.5f);

        // periodic wrap: ((phi + 0.5) mod 1.0) - 0.5, Python-mod semantics
        float ttp = phi + 0.5f;
        ttp -= floorf(ttp);
        const float phip = ttp - 0.5f;

        const float sv = scur[(size_t)b * NN + i];
        const float g  = tanhf(phip) * (1.0f - sv);
        const float ds = DT_F * (gamma[i] * g - sv / tau[i]);
        snxt[(size_t)b * NN + i] = fminf(fmaxf(sv + ds, -1.0f), 1.0f);
    }
}

// ---------------- extract output nodes ----------------
__global__ void soen_extract_kernel(const float* __restrict__ s, float* __restrict__ out) {
    int idx = blockIdx.x * blockDim.x + threadIdx.x; // 0..4095
    int b = idx >> 7;
    int o = idx & 127;
    out[b * NOUT + o] = s[(size_t)b * NN + (NIN + NHID) + o];
}

extern "C" void kernel_launch(void* const* d_in, const int* in_sizes, int n_in,
                              void* d_out, int out_size, void* d_ws, size_t ws_size,
                              hipStream_t stream) {
    (void)in_sizes; (void)n_in; (void)out_size;
    const float* x     = (const float*)d_in[0];   // [32,1024]
    const float* J     = (const float*)d_in[1];   // [4096,4096]
    const float* gamma = (const float*)d_in[2];   // [4096]
    const float* tau   = (const float*)d_in[3];   // [4096]
    const float* flux  = (const float*)d_in[4];   // [4096]
    const float* mask  = (const float*)d_in[5];   // [4096,4096]
    const float* noise = (const float*)d_in[6];   // [100,32,4096]
    float* out = (float*)d_out;

    const size_t sElems = (size_t)NB * NN;                 // 131072 floats
    float* sA = (float*)d_ws;
    float* sB = sA + sElems;
    const size_t jmOff = 2 * sElems * sizeof(float);       // 1 MB
    const bool useJm = ws_size >= jmOff + (size_t)NN * NN * sizeof(float);
    float* Jm = (float*)((char*)d_ws + jmOff);

    // s0 = 0
    soen_zero_kernel<<<(int)((sElems + 255) / 256), 256, 0, stream>>>(sA, (int)sElems);

    // precompute masked weights once if workspace allows (64 MB, L2-resident)
    if (useJm) {
        const int n4 = (NN * NN) / 4;
        soen_maskmul_kernel<<<(n4 + 255) / 256, 256, 0, stream>>>(
            (const float4*)J, (const float4*)mask, (float4*)Jm, n4);
    }

    float* cur = sA;
    float* nxt = sB;
    for (int t = 0; t < 100; ++t) {
        const float* nt = noise + (size_t)t * NB * NN;
        if (useJm) {
            soen_step_kernel<false><<<64, 256, 0, stream>>>(
                cur, nxt, Jm, nullptr, x, flux, gamma, tau, nt);
        } else {
            soen_step_kernel<true><<<64, 256, 0, stream>>>(
                cur, nxt, J, mask, x, flux, gamma, tau, nt);
        }
        float* tmp = cur; cur = nxt; nxt = tmp;
    }

    soen_extract_kernel<<<16, 256, 0, stream>>>(cur, out);
}